// LaplacianPyramid_21827023798920
// MI455X (gfx1250) — compile-verified
//
#include <hip/hip_runtime.h>

typedef __attribute__((ext_vector_type(2))) float v2f;
typedef __attribute__((ext_vector_type(8))) float v8f;

// ---------------------------------------------------------------------------
// Fused 5x5 Gaussian blur (reflect pad 2) + 2x decimation.
// One wave32 computes one 16x16 tile of the H/2 x W/2 output.
// Vertical pass expressed as W(16x36) x H(36x16) via 9 chained
// V_WMMA_F32_16X16X4_F32 ops (fp32 matrix core path on gfx1250).
// ---------------------------------------------------------------------------
__global__ __launch_bounds__(32)
void blur_down_wmma(const float* __restrict__ src, float* __restrict__ dst,
                    int H, int W) {
    const int ho = H >> 1, wo = W >> 1;
    __shared__ float S[36 * 36];               // reflected source tile

    const int tid = threadIdx.x;               // 0..31, one wave
    const int tx0 = blockIdx.x * 16;           // output tile origin (cols)
    const int ty0 = blockIdx.y * 16;           // output tile origin (rows)
    const float* sp = src + (long)blockIdx.z * H * W;
    float*       dp = dst + (long)blockIdx.z * ho * wo;

    // Load 36x36 source tile, rows 2*ty0-2 .. 2*ty0+33, cols 2*tx0-2 .. 2*tx0+33,
    // with jnp.pad 'reflect' semantics: -q for q<0, 2*(L-1)-q for q>=L.
    for (int e = tid; e < 36 * 36; e += 32) {
        int u = e / 36, v = e - u * 36;
        int gr = 2 * ty0 - 2 + u;
        int gc = 2 * tx0 - 2 + v;
        gr = gr < 0 ? -gr : (gr >= H ? 2 * (H - 1) - gr : gr);
        gc = gc < 0 ? -gc : (gc >= W ? 2 * (W - 1) - gc : gc);
        S[e] = sp[(long)gr * W + gc];
    }
    __syncthreads();

    const int s = tid >> 4;    // half-wave: selects K pair {2s, 2s+1} of each 16x4 slab
    const int N = tid & 15;    // B-matrix column / A-matrix row (M) for this lane

    // Horizontal 5-tap at even source columns: this lane needs H[4c+2s+t][N].
    // 1-D weights: {1,4,6,4,1}/16.
    float hv[18];
#pragma unroll
    for (int c = 0; c < 9; ++c) {
#pragma unroll
        for (int t = 0; t < 2; ++t) {
            const int u = 4 * c + 2 * s + t;
            const float* row = &S[u * 36 + 2 * N];
            float acc = 0.0625f * row[0];
            acc += 0.25f   * row[1];
            acc += 0.375f  * row[2];
            acc += 0.25f   * row[3];
            acc += 0.0625f * row[4];
            hv[2 * c + t] = acc;
        }
    }

    // out[y][x] = sum_h Wm[y][h] * H[h][x],  Wm[y][h] = kv[h-2y] if 0<=h-2y<5.
    // A (16x4 slab of Wm) per ISA layout: lanes 0-15 -> K=0,1 ; lanes 16-31 -> K=2,3.
    // d0 = 4c + 2s - 2N is always even, so the weight pair is a branchless select:
    //   a.x (even offset): d0==2 -> 6/16 ; d0 in {0,4} -> 1/16 ; else 0
    //   a.y (odd  offset): d0 in {0,2}   -> 4/16 ; else 0
    v8f acc = {0.f, 0.f, 0.f, 0.f, 0.f, 0.f, 0.f, 0.f};
#pragma unroll
    for (int c = 0; c < 9; ++c) {
        const int d0 = 4 * c + 2 * s - 2 * N;
        v2f a, b;
        a.x = (d0 == 2) ? 0.375f : ((d0 == 0 || d0 == 4) ? 0.0625f : 0.f);
        a.y = (d0 == 0 || d0 == 2) ? 0.25f : 0.f;
        b.x = hv[2 * c];
        b.y = hv[2 * c + 1];
        acc = __builtin_amdgcn_wmma_f32_16x16x4_f32(
            /*neg_a=*/false, a, /*neg_b=*/false, b,
            /*c_mod=*/(short)0, acc, /*reuse_a=*/false, /*reuse_b=*/false);
    }

    // C/D layout: VGPR g holds M=g (lanes 0-15) and M=g+8 (lanes 16-31), N=lane&15.
#pragma unroll
    for (int g = 0; g < 8; ++g) {
        const int M = g + 8 * s;
        dp[(long)(ty0 + M) * wo + (tx0 + N)] = acc[g];
    }
}

// ---------------------------------------------------------------------------
// upsample(x) with 4*kernel on the zero-stuffed grid, fused with elementwise op:
//   add_mode == 0 :  out = a - up(x)      (decompose: laplacian level)
//   add_mode == 1 :  out = up(x) + a      (reconstruct)
// Parity-specialized: even outputs use {1,6,1}/8 per axis, odd use {4,4}/8.
// Reflect on the stuffed grid maps to: q<0 -> -q ; q>=h -> 2h-1-q.
// ---------------------------------------------------------------------------
__global__ __launch_bounds__(256)
void upsample_axpy(const float* __restrict__ x, const float* __restrict__ a,
                   float* __restrict__ out, int h, int w, int total, int add_mode) {
    const int tid = blockIdx.x * blockDim.x + threadIdx.x;
    if (tid >= total) return;

    const int H2 = h * 2, W2 = w * 2;
    const int per = H2 * W2;
    const int img = tid / per;
    int rem = tid - img * per;
    const int Y = rem / W2;
    const int X = rem - Y * W2;

    const float* xp = x + (long)img * h * w;
    const int yc = Y >> 1, xc = X >> 1;

    int ry[3], rx[3], ny, nx;
    float wy[3], wx[3];
    if (Y & 1) { ny = 2; ry[0] = yc;     ry[1] = yc + 1;
                 wy[0] = 0.5f; wy[1] = 0.5f; wy[2] = 0.f; ry[2] = 0; }
    else       { ny = 3; ry[0] = yc - 1; ry[1] = yc; ry[2] = yc + 1;
                 wy[0] = 0.125f; wy[1] = 0.75f; wy[2] = 0.125f; }
    if (X & 1) { nx = 2; rx[0] = xc;     rx[1] = xc + 1;
                 wx[0] = 0.5f; wx[1] = 0.5f; wx[2] = 0.f; rx[2] = 0; }
    else       { nx = 3; rx[0] = xc - 1; rx[1] = xc; rx[2] = xc + 1;
                 wx[0] = 0.125f; wx[1] = 0.75f; wx[2] = 0.125f; }

#pragma unroll
    for (int i = 0; i < 3; ++i) {
        int r = ry[i]; ry[i] = r < 0 ? -r : (r >= h ? 2 * h - 1 - r : r);
        int c = rx[i]; rx[i] = c < 0 ? -c : (c >= w ? 2 * w - 1 - c : c);
    }

    float u = 0.f;
    for (int i = 0; i < ny; ++i) {
        const float* rp = xp + (long)ry[i] * w;
        float hsum = 0.f;
        for (int j = 0; j < nx; ++j) hsum += wx[j] * rp[rx[j]];
        u += wy[i] * hsum;
    }

    const float av = a[tid];
    out[tid] = add_mode ? (u + av) : (av - u);
}

// ---------------------------------------------------------------------------
extern "C" void kernel_launch(void* const* d_in, const int* in_sizes, int n_in,
                              void* d_out, int out_size, void* d_ws, size_t ws_size,
                              hipStream_t stream) {
    const float* img = (const float*)d_in[0];   // (8,3,1024,1024) fp32
    // d_in[1] is the fixed Gaussian kernel; weights are compile-time constants here.
    float* out = (float*)d_out;
    float* ws  = (float*)d_ws;

    const int H0 = 1024;
    const int NC = in_sizes[0] / (H0 * H0);     // B*C = 24 channel-images

    size_t o = 0;
    float* down1 = ws + o; o += (size_t)NC * 512 * 512;
    float* down2 = ws + o; o += (size_t)NC * 256 * 256;
    float* down3 = ws + o; o += (size_t)NC * 128 * 128;
    float* down4 = ws + o; o += (size_t)NC * 64  * 64;
    float* lap0  = ws + o; o += (size_t)NC * 1024 * 1024;
    float* lap1  = ws + o; o += (size_t)NC * 512 * 512;
    float* lap2  = ws + o; o += (size_t)NC * 256 * 256;
    float* lap3  = ws + o; o += (size_t)NC * 128 * 128;
    (void)ws_size; (void)n_in; (void)out_size;

    auto bd = [&](const float* s, float* d, int H) {
        dim3 grid(H / 32, H / 32, NC);          // (H/2)/16 tiles per axis
        blur_down_wmma<<<grid, dim3(32, 1, 1), 0, stream>>>(s, d, H, H);
    };
    auto ua = [&](const float* x, const float* a, float* op, int h, int add) {
        const int total = NC * 4 * h * h;
        upsample_axpy<<<(total + 255) / 256, 256, 0, stream>>>(x, a, op, h, h, total, add);
    };

    // Decompose: down_{k+1} = decimate(blur(cur_k)); lap_k = cur_k - up(down_{k+1})
    bd(img,   down1, 1024);
    bd(down1, down2, 512);
    bd(down2, down3, 256);
    bd(down3, down4, 128);
    ua(down1, img,   lap0, 512, 0);
    ua(down2, down1, lap1, 256, 0);
    ua(down3, down2, lap2, 128, 0);
    ua(down4, down3, lap3, 64,  0);

    // Reconstruct: image_k = up(image_{k+1}) + lap_k  (in-place into lap buffers)
    ua(down4, lap3, lap3, 64,  1);   // image3
    ua(lap3,  lap2, lap2, 128, 1);   // image2
    ua(lap2,  lap1, lap1, 256, 1);   // image1
    ua(lap1,  lap0, out,  512, 1);   // image0 -> d_out
}